// Centroids_20615843021281
// MI455X (gfx1250) — compile-verified
//
#include <hip/hip_runtime.h>
#include <hip/hip_bf16.h>
#include <math.h>

#define N_BANK 200000
#define DIM    512
#define BATCH  8192
#define YC     10
#define KC     8
#define YK     (YC * KC)   // 80
#define MOM    0.1f

typedef __attribute__((ext_vector_type(2))) float v2f;
typedef __attribute__((ext_vector_type(4))) float f4;
typedef __attribute__((ext_vector_type(8))) float v8f;

// ---------------------------------------------------------------------------
// Pass 1: base copies of the small per-id state arrays + winner init.
// ---------------------------------------------------------------------------
__global__ void k_init(const int* __restrict__ assigns,
                       const int* __restrict__ corrects,
                       const float* __restrict__ losses,
                       float* __restrict__ assignsOut,
                       float* __restrict__ correctsOut,
                       float* __restrict__ lossesOut,
                       int* __restrict__ winner) {
  int i = blockIdx.x * blockDim.x + threadIdx.x;
  if (i < N_BANK) {
    winner[i]      = -1;
    assignsOut[i]  = (float)assigns[i];
    correctsOut[i] = (float)corrects[i];
    lossesOut[i]   = losses[i];
  }
}

// ---------------------------------------------------------------------------
// Pass 2: streaming copy of the 400MB feature bank (dominant cost, ~35us at
// 23.3 TB/s). Non-temporal so the bank doesn't thrash L2; the 16MB
// feature_new working set and cluster means stay resident instead.
// ---------------------------------------------------------------------------
__global__ void k_copy_bank(const f4* __restrict__ src, f4* __restrict__ dst,
                            long total4) {
  long i      = (long)blockIdx.x * blockDim.x + threadIdx.x;
  long stride = (long)gridDim.x * blockDim.x;
  for (; i < total4; i += stride) {
    f4 v = __builtin_nontemporal_load(&src[i]);
    __builtin_nontemporal_store(v, &dst[i]);
  }
}

// ---------------------------------------------------------------------------
// Pass 3: deterministic duplicate resolution: highest batch index wins.
// ---------------------------------------------------------------------------
__global__ void k_vote(const int* __restrict__ ids, int* __restrict__ winner) {
  int b = blockIdx.x * blockDim.x + threadIdx.x;
  if (b < BATCH) atomicMax(&winner[ids[b]], b);
}

// ---------------------------------------------------------------------------
// Pass 4: per-row L1 normalize, EMA with old bank row, L1 normalize again.
// One wave32 per row (512 f32 = 4 float4 per lane). Wave reduction via
// __shfl_xor over 32 lanes (wave32!). Writes feature_new to workspace and
// scatters the winning rows into the output bank.
// ---------------------------------------------------------------------------
__global__ void k_feat(const float* __restrict__ feature,
                       const float* __restrict__ bank,
                       const int* __restrict__ ids,
                       const int* __restrict__ winner,
                       float* __restrict__ fnw,
                       float* __restrict__ bankOut) {
  int wave = (int)((blockIdx.x * blockDim.x + threadIdx.x) >> 5);
  int lane = threadIdx.x & 31;
  if (wave >= BATCH) return;

  const f4* frow = (const f4*)(feature + (long)wave * DIM);
  f4 x[4];
  float s = 0.f;
#pragma unroll
  for (int j = 0; j < 4; ++j) {
    x[j] = frow[lane + 32 * j];
    s += fabsf(x[j].x) + fabsf(x[j].y) + fabsf(x[j].z) + fabsf(x[j].w);
  }
#pragma unroll
  for (int off = 16; off > 0; off >>= 1) s += __shfl_xor(s, off, 32);
  float inv1 = 1.0f / fmaxf(s, 1e-12f);

  int id = ids[wave];
  const f4* brow = (const f4*)(bank + (long)id * DIM);
  f4 v[4];
  float s2 = 0.f;
#pragma unroll
  for (int j = 0; j < 4; ++j) {
    f4 bb = brow[lane + 32 * j];
    v[j].x = (1.0f - MOM) * bb.x + MOM * (x[j].x * inv1);
    v[j].y = (1.0f - MOM) * bb.y + MOM * (x[j].y * inv1);
    v[j].z = (1.0f - MOM) * bb.z + MOM * (x[j].z * inv1);
    v[j].w = (1.0f - MOM) * bb.w + MOM * (x[j].w * inv1);
    s2 += fabsf(v[j].x) + fabsf(v[j].y) + fabsf(v[j].z) + fabsf(v[j].w);
  }
#pragma unroll
  for (int off = 16; off > 0; off >>= 1) s2 += __shfl_xor(s2, off, 32);
  float inv2 = 1.0f / fmaxf(s2, 1e-12f);

  f4* fout = (f4*)(fnw + (long)wave * DIM);
  bool win = (winner[id] == wave);
  f4* bout = (f4*)(bankOut + (long)id * DIM);
#pragma unroll
  for (int j = 0; j < 4; ++j) {
    f4 r = v[j] * inv2;
    fout[lane + 32 * j] = r;
    if (win) bout[lane + 32 * j] = r;
  }
}

// ---------------------------------------------------------------------------
// Pass 5: sim[yk, b] = cluster_means[yk,:] . feature_new[b,:]
// Full-precision f32 WMMA: V_WMMA_F32_16X16X4_F32, one 16x16 tile per wave,
// K accumulated 4 at a time over 512. M tiles = 80/16 = 5, N tiles = 512.
// A layout: lane%16 = M row, lane-half + 2 VGPRs = K 0..3.
// B layout: lane%16 = N col, lane-half + 2 VGPRs = K 0..3.
// ---------------------------------------------------------------------------
__global__ void k_wmma(const float* __restrict__ cm,
                       const float* __restrict__ fnw,
                       float* __restrict__ sim) {
  int w    = (int)((blockIdx.x * blockDim.x + threadIdx.x) >> 5);
  int lane = threadIdx.x & 31;
  const int NTILES = BATCH / 16;                 // 512
  if (w >= (YK / 16) * NTILES) return;
  int m = w / NTILES;                            // 0..4
  int n = w % NTILES;                            // 0..511
  int idx   = lane & 15;
  int khalf = (lane < 16) ? 0 : 2;

  const float* arow = cm  + (long)(m * 16 + idx) * DIM + khalf;
  const float* brow = fnw + (long)(n * 16 + idx) * DIM + khalf;

  v8f acc = {};
#pragma unroll 4
  for (int k = 0; k < DIM; k += 4) {
    v2f a = *(const v2f*)(arow + k);
    v2f b = *(const v2f*)(brow + k);
    acc = __builtin_amdgcn_wmma_f32_16x16x4_f32(
        /*neg_a=*/false, a, /*neg_b=*/false, b,
        /*c_mod=*/(short)0, acc, /*reuse_a=*/false, /*reuse_b=*/false);
  }

  int rowbase = m * 16 + ((lane < 16) ? 0 : 8);
  int col     = n * 16 + idx;
#pragma unroll
  for (int r = 0; r < 8; ++r)
    sim[(long)(rowbase + r) * BATCH + col] = acc[r];
}

// ---------------------------------------------------------------------------
// Pass 6: masked argmax over 80 clusters (subtract 1e4 for wrong class,
// first-max-wins like jnp.argmax), out-argmax correctness flag, per-sample
// cross entropy; scatter winners.
// ---------------------------------------------------------------------------
__global__ void k_post(const float* __restrict__ sim,
                       const float* __restrict__ out,
                       const int* __restrict__ target,
                       const int* __restrict__ ids,
                       const int* __restrict__ winner,
                       float* __restrict__ assignsOut,
                       float* __restrict__ correctsOut,
                       float* __restrict__ lossesOut,
                       float* __restrict__ newAssignsOut) {
  int b = blockIdx.x * blockDim.x + threadIdx.x;
  if (b >= BATCH) return;
  int t = target[b];

  float best = -INFINITY;
  int besti = 0;
  for (int yk = 0; yk < YK; ++yk) {
    float sv = sim[(long)yk * BATCH + b];
    if ((yk >> 3) != t) sv -= 10000.0f;          // class = yk / KC, KC == 8
    if (sv > best) { best = sv; besti = yk; }    // strict > : first max wins
  }
  newAssignsOut[b] = (float)besti;

  const float* orow = out + (long)b * YC;
  float mx = orow[0];
  int am = 0;
  for (int c = 1; c < YC; ++c) {
    float v = orow[c];
    if (v > mx) { mx = v; am = c; }
  }
  float S = 0.f;
  for (int c = 0; c < YC; ++c) S += expf(orow[c] - mx);
  float ce = -(orow[t] - mx - logf(S));

  int id = ids[b];
  if (winner[id] == b) {
    assignsOut[id]  = (float)besti;
    correctsOut[id] = (am == t) ? 1.0f : 0.0f;
    lossesOut[id]   = ce;
  }
}

// ---------------------------------------------------------------------------
extern "C" void kernel_launch(void* const* d_in, const int* in_sizes, int n_in,
                              void* d_out, int out_size, void* d_ws, size_t ws_size,
                              hipStream_t stream) {
  (void)in_sizes; (void)n_in; (void)out_size; (void)ws_size;

  const float* feature  = (const float*)d_in[0];   // [B, D]
  const float* out      = (const float*)d_in[1];   // [B, Y]
  const float* bank     = (const float*)d_in[2];   // [N, D]
  const float* cm       = (const float*)d_in[3];   // [Y, K, D] = [80, 512]
  const float* losses   = (const float*)d_in[4];   // [N]
  const int*   target   = (const int*)d_in[5];     // [B]
  const int*   ids      = (const int*)d_in[6];     // [B]
  const int*   assigns  = (const int*)d_in[7];     // [N]
  const int*   corrects = (const int*)d_in[8];     // [N]

  float* o           = (float*)d_out;
  float* bankOut     = o;                                    // N*D
  float* assignsOut  = o + (long)N_BANK * DIM;               // N
  float* correctsOut = assignsOut + N_BANK;                  // N
  float* lossesOut   = correctsOut + N_BANK;                 // N
  float* newAssigns  = lossesOut + N_BANK;                   // B

  char* ws     = (char*)d_ws;
  float* fnw   = (float*)ws;                                 // B*D f32 = 16 MB
  float* simB  = (float*)(ws + (size_t)BATCH * DIM * 4);     // 80*B f32 = 2.56 MB
  int*   winner = (int*)(ws + (size_t)BATCH * DIM * 4 + (size_t)YK * BATCH * 4); // N i32

  // 1. base copies + winner init
  k_init<<<(N_BANK + 255) / 256, 256, 0, stream>>>(
      assigns, corrects, losses, assignsOut, correctsOut, lossesOut, winner);

  // 2. streaming bank copy (dominant: ~820 MB of HBM traffic)
  long total4 = (long)N_BANK * DIM / 4;
  k_copy_bank<<<4096, 256, 0, stream>>>((const f4*)bank, (f4*)bankOut, total4);

  // 3. duplicate-id winner vote
  k_vote<<<(BATCH + 255) / 256, 256, 0, stream>>>(ids, winner);

  // 4. feature_new compute + bank row scatter (one wave32 per row)
  k_feat<<<BATCH / 4, 128, 0, stream>>>(feature, bank, ids, winner, fnw, bankOut);

  // 5. WMMA f32 GEMM: 5 x 512 tiles of 16x16, 4 waves per block
  k_wmma<<<((YK / 16) * (BATCH / 16)) / 4, 128, 0, stream>>>(cm, fnw, simB);

  // 6. masked argmax + CE + scatters
  k_post<<<BATCH / 256, 256, 0, stream>>>(simB, out, target, ids, winner,
                                          assignsOut, correctsOut, lossesOut,
                                          newAssigns);
}